// InnerProductConv_1159641170488
// MI455X (gfx1250) — compile-verified
//
#include <hip/hip_runtime.h>

// ----------------------------------------------------------------------------
// InnerProductConv refactored:
//   agg[d] = ( sum_{e: dst(e)=d} src[edge_src[e]] ) @ dst[d] / sqrt(P)
// Phase 1: zero output, scatter-add gathered src rows (atomic f32, L2-resident)
// Phase 2: per-node 8x8 @ 8x8 matmul, 2 nodes per wave via V_WMMA_F32_16X16X4_F32
// ----------------------------------------------------------------------------

typedef __attribute__((ext_vector_type(2))) float v2f;
typedef __attribute__((ext_vector_type(8))) float v8f;

__global__ void zero_f32(float* __restrict__ p, int n) {
    int t = blockIdx.x * blockDim.x + threadIdx.x;
    if (t < n) p[t] = 0.0f;
}

// 16 threads per edge, each handling 4 contiguous floats of the 64-float row.
__global__ void scatter_src_sum(const float* __restrict__ src_feat,
                                const int* __restrict__ edge_src,
                                const int* __restrict__ edge_dst,
                                float* __restrict__ accum,
                                int n_edges) {
    int t = blockIdx.x * blockDim.x + threadIdx.x;
    int e = t >> 4;
    if (e >= n_edges) return;
    int j = (t & 15) << 2;
    int s = edge_src[e];
    int d = edge_dst[e];
    const float4 v = *(const float4*)(src_feat + s * 64 + j);
    float* o = accum + d * 64 + j;
    atomicAdd(o + 0, v.x);
    atomicAdd(o + 1, v.y);
    atomicAdd(o + 2, v.z);
    atomicAdd(o + 3, v.w);
}

// One wave computes C = [S(d0);S(d1)](16x8) @ [B(d0) B(d1)](8x16) via two
// K=4 fp32 WMMA steps; diagonal 8x8 blocks are the per-node results.
__global__ void __launch_bounds__(256)
node_pair_wmma(const float* __restrict__ dst_feat,
               float* __restrict__ out,
               int n_nodes) {
    const int lane = threadIdx.x & 31;
    const int wave = (int)((blockIdx.x * blockDim.x + threadIdx.x) >> 5);
    int d0 = 2 * wave;
    int d1 = 2 * wave + 1;
    if (d0 >= n_nodes) return;          // wave-uniform (taken by surplus waves of the last block)
    if (d1 >= n_nodes) d1 = d0;         // odd tail: duplicate node, benign same-value write

    const int half = lane >> 4;         // selects K sub-pair within each WMMA chunk
    const int l    = lane & 15;
    const int n8   = l & 7;
    const int node = (l < 8) ? d0 : d1; // row/col owner for A/B operand lanes

    // A operand: A_big[M=l][k] from accumulated S tiles (currently living in `out`)
    //   lane<16 holds k = kbase+{0,1}; lane>=16 holds k = kbase+{2,3}
    const float* abase = out + node * 64 + n8 * 8 + 2 * half;   // 8B aligned
    const float2 a0f = *(const float2*)(abase + 0);             // kbase = 0
    const float2 a1f = *(const float2*)(abase + 4);             // kbase = 4
    v2f a0; a0.x = a0f.x; a0.y = a0f.y;
    v2f a1; a1.x = a1f.x; a1.y = a1f.y;

    // B operand: B_big[k][N=l] from dst_feat tiles (row-major 8x8 => stride 8)
    const float* bbase = dst_feat + node * 64 + n8 + 16 * half;
    v2f b0; b0.x = bbase[0];  b0.y = bbase[8];                  // k rows kbase+{0,1}
    v2f b1; b1.x = bbase[32]; b1.y = bbase[40];                 // k rows 4+{0,1}

    v8f c = {};
    c = __builtin_amdgcn_wmma_f32_16x16x4_f32(false, a0, false, b0,
                                              (short)0, c, false, false);
    c = __builtin_amdgcn_wmma_f32_16x16x4_f32(false, a1, false, b1,
                                              (short)0, c, false, false);

    // C layout: VGPR r, lanes 0-15 -> C[r][l]; lanes 16-31 -> C[r+8][l].
    // Keep diagonal blocks: lanes 0-7 (node d0), lanes 24-31 (node d1).
    const float scale = 0.35355339059327379f;  // 1/sqrt(8)
    const bool w0 = (half == 0) && (l < 8);
    const bool w1 = (half == 1) && (l >= 8);
    if (w0 || w1) {
        float* orow = out + (w0 ? d0 : d1) * 64 + n8;
#pragma unroll
        for (int r = 0; r < 8; ++r) orow[r * 8] = c[r] * scale;
    }
}

extern "C" void kernel_launch(void* const* d_in, const int* in_sizes, int n_in,
                              void* d_out, int out_size, void* d_ws, size_t ws_size,
                              hipStream_t stream) {
    const float* src_feat = (const float*)d_in[0];
    const float* dst_feat = (const float*)d_in[1];
    const int*   edge_src = (const int*)d_in[2];
    const int*   edge_dst = (const int*)d_in[3];
    float* out = (float*)d_out;

    const int n_edges = in_sizes[2];
    const int n_nodes = out_size / 64;

    // Phase 0: zero the accumulator/output (harness poisons d_out).
    zero_f32<<<(out_size + 255) / 256, 256, 0, stream>>>(out, out_size);

    // Phase 1: scatter-add src rows keyed by edge_dst.
    const long long t1 = (long long)n_edges * 16;
    scatter_src_sum<<<(unsigned)((t1 + 255) / 256), 256, 0, stream>>>(
        src_feat, edge_src, edge_dst, out, n_edges);

    // Phase 2: per-node matmul, 2 nodes per wave, 8 waves per block.
    const int pairs = (n_nodes + 1) / 2;
    node_pair_wmma<<<(pairs + 7) / 8, 256, 0, stream>>>(dst_feat, out, n_nodes);
}